// VectorGraphicsRGBPaddedAccelIntegrandSlang_5222680232016
// MI455X (gfx1250) — compile-verified
//
#include <hip/hip_runtime.h>
#include <hip/hip_bf16.h>
#include <stdint.h>

#define GRID_G   64
#define MAXE     32
#define NPRIM    900
#define IMG_N    (2048 * 2048)

// Workspace layout (bytes), all 16B-aligned
#define WS_PRIMS    0        // 900 * 48  = 43200
#define WS_AABB     43264    // 900 * 16  = 14400  -> ends 57664
#define WS_COUNTS   57728    // 4096 * 4  = 16384  -> ends 74112
#define WS_ENTRIES  74112    // 4096*32*4 = 524288 -> ends 598400

#define PRIMS_BYTES   43200
#define COUNTS_BYTES  16384

typedef float f2 __attribute__((ext_vector_type(2)));

// -------------------------------------------------------------------------
// Kernel 1: pack per-primitive invariants + width-expanded AABBs.
//   prims[3p+0] = {a0x, a0y, dx, dy}
//   prims[3p+1] = {max(dd,1e-12), w*w, opacity, 0}
//   prims[3p+2] = {r, g, b, 0}
// -------------------------------------------------------------------------
__global__ void prep_prims(const float* __restrict__ cp,
                           const float* __restrict__ sw,
                           const float* __restrict__ fc,
                           const float* __restrict__ op,
                           float4* __restrict__ prims,
                           float4* __restrict__ aabb) {
    int p = blockIdx.x * blockDim.x + threadIdx.x;
    if (p >= NPRIM) return;
    float x0 = cp[6 * p + 0], y0 = cp[6 * p + 1];
    float x1 = cp[6 * p + 2], y1 = cp[6 * p + 3];
    float w  = sw[p];
    float dx = x1 - x0, dy = y1 - y0;
    float dd = fmaxf(dx * dx + dy * dy, 1e-12f);
    prims[3 * p + 0] = make_float4(x0, y0, dx, dy);
    prims[3 * p + 1] = make_float4(dd, w * w, op[p], 0.0f);
    prims[3 * p + 2] = make_float4(fc[3 * p + 0], fc[3 * p + 1], fc[3 * p + 2], 0.0f);
    aabb[p] = make_float4(fminf(x0, x1) - w,   // xmin
                          fmaxf(x0, x1) + w,   // xmax
                          fminf(y0, y1) - w,   // ymin
                          fmaxf(y0, y1) + w);  // ymax
}

// -------------------------------------------------------------------------
// Kernel 2: per-cell acceleration grid. One thread = one cell.
// Records the first <=32 overlapping primitive indices in ascending order,
// which matches the reference's argsort(~mask)[:, :32] + min(count, 32).
// -------------------------------------------------------------------------
__global__ __launch_bounds__(256) void build_grid(const float4* __restrict__ aabb,
                                                  int* __restrict__ counts,
                                                  int* __restrict__ entries) {
    __shared__ float4 sa[NPRIM];
    for (int i = threadIdx.x; i < NPRIM; i += blockDim.x) sa[i] = aabb[i];
    __syncthreads();

    int cell = blockIdx.x * blockDim.x + threadIdx.x;   // 16 blocks * 256 = 4096
    int cx = cell >> 6;        // X index (matches cell = ix*G + iy)
    int cy = cell & 63;        // Y index
    const float inv = 1.0f / (float)GRID_G;             // 1/64, exact
    float lox = cx * inv, hix = (cx + 1) * inv;
    float loy = cy * inv, hiy = (cy + 1) * inv;

    int cnt = 0;
    int* e = entries + cell * MAXE;
    for (int p = 0; p < NPRIM; ++p) {
        float4 b = sa[p];
        // ox = (xmin < hi) & (xmax >= lo);  oy likewise
        bool ov = (b.x < hix) & (b.y >= lox) & (b.z < hiy) & (b.w >= loy);
        if (ov && cnt < MAXE) { e[cnt] = p; ++cnt; }
    }
    counts[cell] = cnt;
}

// -------------------------------------------------------------------------
// Kernel 3: shading. Persistent blocks; each block first stages the packed
// primitive table (43200 B) and the counts table (16384 B) into LDS with
// CDNA5 async global->LDS loads (ASYNCcnt), then grid-strides over samples.
// Sample coordinates are streamed with a non-temporal hint so the 32 MB
// single-use stream does not displace the gather tables in WGP$/L2.
// -------------------------------------------------------------------------
__global__ __launch_bounds__(256) void shade(const f2*     __restrict__ xs,
                                             const float*  __restrict__ bg,
                                             const float4* __restrict__ gprims,
                                             const int*    __restrict__ gcounts,
                                             const int*    __restrict__ entries,
                                             float* __restrict__ out,
                                             int total) {
    __shared__ __align__(16) unsigned char smem[PRIMS_BYTES + COUNTS_BYTES];

    // Low 32 bits of a flat LDS address are the wave-relative LDS byte offset
    // (the address truncated to 32 bits); async loads take the LDS offset in VDST.
    const unsigned ldsBase = (unsigned)(uintptr_t)(&smem[0]);

    {
        const uint64_t psrc = (uint64_t)(uintptr_t)gprims;
        for (int c = threadIdx.x; c < PRIMS_BYTES / 16; c += 256) {
            unsigned lAddr = ldsBase + (unsigned)c * 16u;
            unsigned gOff  = (unsigned)c * 16u;
            asm volatile("global_load_async_to_lds_b128 %0, %1, %2"
                         :: "v"(lAddr), "v"(gOff), "s"(psrc) : "memory");
        }
        const uint64_t csrc = (uint64_t)(uintptr_t)gcounts;
        for (int c = threadIdx.x; c < COUNTS_BYTES / 16; c += 256) {
            unsigned lAddr = ldsBase + (unsigned)PRIMS_BYTES + (unsigned)c * 16u;
            unsigned gOff  = (unsigned)c * 16u;
            asm volatile("global_load_async_to_lds_b128 %0, %1, %2"
                         :: "v"(lAddr), "v"(gOff), "s"(csrc) : "memory");
        }
    }
    asm volatile("s_wait_asynccnt 0" ::: "memory");
    __syncthreads();

    const float4* sprims  = (const float4*)smem;
    const int*    scounts = (const int*)(smem + PRIMS_BYTES);

    const float bgr = bg[0], bgg = bg[1], bgb = bg[2];
    const int stride = gridDim.x * blockDim.x;

    for (int i = blockIdx.x * blockDim.x + threadIdx.x; i < total; i += stride) {
        f2 p = __builtin_nontemporal_load(&xs[i]);   // global_load_b64 th:TH_NT
        int ix = min(max((int)(p.x * 64.0f), 0), 63);
        int iy = min(max((int)(p.y * 64.0f), 0), 63);
        int cell = (ix << 6) | iy;
        int cnt = scounts[cell];
        const int* e = entries + cell * MAXE;

        float cr = bgr, cg = bgg, cb = bgb;
        for (int s = 0; s < cnt; ++s) {
            int prim = e[s];
            float4 A = sprims[prim * 3 + 0];   // a0x a0y dx dy
            float4 B = sprims[prim * 3 + 1];   // ddmax, w^2, opacity, -
            float relx = p.x - A.x, rely = p.y - A.y;
            float t = (relx * A.z + rely * A.w) / B.x;
            t = fminf(fmaxf(t, 0.0f), 1.0f);
            float exd = relx - t * A.z, eyd = rely - t * A.w;
            float d2 = exd * exd + eyd * eyd;
            if (d2 <= B.y) {                   // dist <= w  (both nonnegative)
                float4 C = sprims[prim * 3 + 2];
                float a = B.z, ia = 1.0f - a;
                cr = cr * ia + C.x * a;
                cg = cg * ia + C.y * a;
                cb = cb * ia + C.z * a;
            }
        }
        out[3 * i + 0] = cr;
        out[3 * i + 1] = cg;
        out[3 * i + 2] = cb;
    }
}

// -------------------------------------------------------------------------
extern "C" void kernel_launch(void* const* d_in, const int* in_sizes, int n_in,
                              void* d_out, int out_size, void* d_ws, size_t ws_size,
                              hipStream_t stream) {
    const float* x   = (const float*)d_in[0];   // (4M, 2)
    const float* cp  = (const float*)d_in[1];   // (900, 6)
    const float* sw  = (const float*)d_in[2];   // (900,)
    const float* fc  = (const float*)d_in[3];   // (900, 3)
    const float* op  = (const float*)d_in[4];   // (900,)
    const float* bg  = (const float*)d_in[5];   // (3,)

    char* ws = (char*)d_ws;
    float4* prims   = (float4*)(ws + WS_PRIMS);
    float4* aabb    = (float4*)(ws + WS_AABB);
    int*    counts  = (int*)   (ws + WS_COUNTS);
    int*    entries = (int*)   (ws + WS_ENTRIES);

    prep_prims<<<(NPRIM + 255) / 256, 256, 0, stream>>>(cp, sw, fc, op, prims, aabb);
    build_grid<<<(GRID_G * GRID_G) / 256, 256, 0, stream>>>(aabb, counts, entries);
    shade<<<4096, 256, 0, stream>>>((const f2*)x, bg, prims, counts, entries,
                                    (float*)d_out, IMG_N);
}